// Net_22600117911898
// MI455X (gfx1250) — compile-verified
//
#include <hip/hip_runtime.h>
#include <math.h>

typedef __attribute__((ext_vector_type(2))) float v2f;
typedef __attribute__((ext_vector_type(8))) float v8f;

#define N_NODES 100000
#define N_EDGES 1600000
#define FEAT    256
#define HIDDEN  128
#define CLASSES 40

// -------------------------------------------------------------------------
// GEMM1: H0 = X @ W1.  X:[N,256] row-major, W1:[256,128] row-major.
// One wave per 16-row tile; wave keeps the full 16x128 output strip in
// registers (8 x v8f accumulators) so X streams from HBM exactly once and
// W1 (128KB) stays cache-resident.
// A-operand layout (ISA 7.12.2, 32-bit A 16x4): lanes 0-15 carry K=0/1 in
// VGPR0/1 for row M=lane; lanes 16-31 carry K=2/3 for row M=lane-16.
// -------------------------------------------------------------------------
__global__ __launch_bounds__(32) void gemm1_wmma(const float* __restrict__ X,
                                                 const float* __restrict__ W1,
                                                 float* __restrict__ H0) {
    const int rt   = blockIdx.x;          // 16-row tile index
    const int lane = threadIdx.x & 31;
    const int half = lane >> 4;           // 0: lanes 0-15, 1: lanes 16-31
    const int l16  = lane & 15;
    const int row  = rt * 16 + l16;
    const int koff = half * 2;

    v8f zero;
#pragma unroll
    for (int r = 0; r < 8; ++r) zero[r] = 0.0f;
    v8f acc[8];
#pragma unroll
    for (int j = 0; j < 8; ++j) acc[j] = zero;

    const float* arow = X + row * FEAT;
    for (int k = 0; k < FEAT; k += 4) {
        v2f a;
        a.x = arow[k + koff + 0];
        a.y = arow[k + koff + 1];
#pragma unroll
        for (int j = 0; j < 8; ++j) {
            const float* bp = W1 + (k + koff) * HIDDEN + j * 16 + l16;
            v2f b;
            b.x = bp[0];
            b.y = bp[HIDDEN];
            acc[j] = __builtin_amdgcn_wmma_f32_16x16x4_f32(
                false, a, false, b, (short)0, acc[j], false, false);
        }
    }
    // D layout: VGPR r -> row (r + 8*half), col = lane&15  (per 16x16 f32 C/D map)
#pragma unroll
    for (int j = 0; j < 8; ++j) {
#pragma unroll
        for (int r = 0; r < 8; ++r) {
            const int orow = rt * 16 + half * 8 + r;
            const int ocol = j * 16 + l16;
            H0[orow * HIDDEN + ocol] = acc[j][r];
        }
    }
}

// -------------------------------------------------------------------------
// SpMM1: AGG1 += edge_vals * H0[edge_src], scattered to edge_dst.
// One wave per edge (edge index uniform across the wave -> scalar loads),
// each lane handles 4 consecutive features via float4 gather + 4 f32 atomics.
// h0 / agg1 are L2-resident (51MB each vs 192MB L2).
// -------------------------------------------------------------------------
__global__ __launch_bounds__(256) void spmm1(const int*   __restrict__ esrc,
                                             const int*   __restrict__ edst,
                                             const float* __restrict__ eval,
                                             const float* __restrict__ H0,
                                             float*       __restrict__ AGG1) {
    const int tid = blockIdx.x * 256 + threadIdx.x;
    const int e   = tid >> 5;              // wave-uniform
    if (e >= N_EDGES) return;
    const int d   = (tid & 31) * 4;

    const int   s = esrc[e];
    const int   t = edst[e];
    const float v = eval[e];

    const float4 hv = *(const float4*)(H0 + s * HIDDEN + d);
    float* o = AGG1 + t * HIDDEN + d;
    atomicAdd(o + 0, v * hv.x);
    atomicAdd(o + 1, v * hv.y);
    atomicAdd(o + 2, v * hv.z);
    atomicAdd(o + 3, v * hv.w);
}

// -------------------------------------------------------------------------
// GEMM2: H2 = relu(AGG1) @ W2.  AGG1:[N,128], W2:[128,40].
// Same WMMA scheme; relu fused into the A-operand load; 40 cols padded to
// 3x16 tiles with branch-free zero-fill on B and guarded stores.
// -------------------------------------------------------------------------
__global__ __launch_bounds__(32) void gemm2_wmma(const float* __restrict__ A1,
                                                 const float* __restrict__ W2,
                                                 float* __restrict__ H2) {
    const int rt   = blockIdx.x;
    const int lane = threadIdx.x & 31;
    const int half = lane >> 4;
    const int l16  = lane & 15;
    const int row  = rt * 16 + l16;
    const int koff = half * 2;

    v8f zero;
#pragma unroll
    for (int r = 0; r < 8; ++r) zero[r] = 0.0f;
    v8f acc[3];
#pragma unroll
    for (int j = 0; j < 3; ++j) acc[j] = zero;

    const float* arow = A1 + row * HIDDEN;
    for (int k = 0; k < HIDDEN; k += 4) {
        v2f a;
        a.x = fmaxf(arow[k + koff + 0], 0.0f);   // fused relu
        a.y = fmaxf(arow[k + koff + 1], 0.0f);
#pragma unroll
        for (int j = 0; j < 3; ++j) {
            const int  col = j * 16 + l16;
            const bool ok  = (col < CLASSES);
            const int  c   = ok ? col : 0;
            const float* bp = W2 + (k + koff) * CLASSES + c;
            v2f b;
            b.x = ok ? bp[0]       : 0.0f;
            b.y = ok ? bp[CLASSES] : 0.0f;
            acc[j] = __builtin_amdgcn_wmma_f32_16x16x4_f32(
                false, a, false, b, (short)0, acc[j], false, false);
        }
    }
#pragma unroll
    for (int j = 0; j < 3; ++j) {
#pragma unroll
        for (int r = 0; r < 8; ++r) {
            const int orow = rt * 16 + half * 8 + r;
            const int ocol = j * 16 + l16;
            if (ocol < CLASSES) H2[orow * CLASSES + ocol] = acc[j][r];
        }
    }
}

// -------------------------------------------------------------------------
// SpMM2: AGG2 += edge_vals * H2[edge_src] scattered to edge_dst, 40 feats.
// One thread per (edge, class). h2/agg2 are 16MB each -> L2-resident.
// -------------------------------------------------------------------------
__global__ __launch_bounds__(256) void spmm2(const int*   __restrict__ esrc,
                                             const int*   __restrict__ edst,
                                             const float* __restrict__ eval,
                                             const float* __restrict__ H2,
                                             float*       __restrict__ AGG2) {
    const int tid = blockIdx.x * 256 + threadIdx.x;
    if (tid >= N_EDGES * CLASSES) return;
    const int e = tid / CLASSES;
    const int d = tid - e * CLASSES;
    const int   s = esrc[e];
    const int   t = edst[e];
    const float v = eval[e];
    atomicAdd(AGG2 + t * CLASSES + d, v * H2[s * CLASSES + d]);
}

// -------------------------------------------------------------------------
// Row-wise log_softmax over 40 classes; one wave per row, shuffle reductions.
// Lane l covers class l (always valid) and class l+32 (valid for l<8).
// -------------------------------------------------------------------------
__global__ __launch_bounds__(256) void logsoftmax_k(const float* __restrict__ AGG2,
                                                    float* __restrict__ OUT) {
    const int wid  = threadIdx.x >> 5;
    const int lane = threadIdx.x & 31;
    const int row  = blockIdx.x * 8 + wid;
    if (row >= N_NODES) return;

    const float* p = AGG2 + row * CLASSES;
    const float a = p[lane];
    const float b = (lane < CLASSES - 32) ? p[32 + lane] : -INFINITY;

    float m = fmaxf(a, b);
#pragma unroll
    for (int off = 16; off; off >>= 1) m = fmaxf(m, __shfl_xor(m, off, 32));

    float s = __expf(a - m) + ((lane < CLASSES - 32) ? __expf(b - m) : 0.0f);
#pragma unroll
    for (int off = 16; off; off >>= 1) s += __shfl_xor(s, off, 32);

    const float l = m + __logf(s);
    float* o = OUT + row * CLASSES;
    o[lane] = a - l;
    if (lane < CLASSES - 32) o[32 + lane] = b - l;
}

extern "C" void kernel_launch(void* const* d_in, const int* in_sizes, int n_in,
                              void* d_out, int out_size, void* d_ws, size_t ws_size,
                              hipStream_t stream) {
    const float* x    = (const float*)d_in[0];
    const float* w1   = (const float*)d_in[1];
    const float* w2   = (const float*)d_in[2];
    const int*   esrc = (const int*)d_in[3];
    const int*   edst = (const int*)d_in[4];
    const float* evl  = (const float*)d_in[5];
    float*       out  = (float*)d_out;

    // Workspace layout (floats): h0[N*128] | agg1[N*128] | h2[N*40] | agg2[N*40]
    float* h0   = (float*)d_ws;
    float* agg1 = h0   + (size_t)N_NODES * HIDDEN;
    float* h2   = agg1 + (size_t)N_NODES * HIDDEN;
    float* agg2 = h2   + (size_t)N_NODES * CLASSES;

    // Accumulators must be zero every call (graph-capture safe memsets).
    hipMemsetAsync(agg1, 0, (size_t)N_NODES * HIDDEN  * sizeof(float), stream);
    hipMemsetAsync(agg2, 0, (size_t)N_NODES * CLASSES * sizeof(float), stream);

    gemm1_wmma<<<N_NODES / 16, 32, 0, stream>>>(x, w1, h0);                       // 6250 waves
    spmm1<<<(N_EDGES * 32) / 256, 256, 0, stream>>>(esrc, edst, evl, h0, agg1);   // 200000 blocks
    gemm2_wmma<<<N_NODES / 16, 32, 0, stream>>>(agg1, w2, h2);
    spmm2<<<(N_EDGES * CLASSES + 255) / 256, 256, 0, stream>>>(esrc, edst, evl, h2, agg2);
    logsoftmax_k<<<(N_NODES + 7) / 8, 256, 0, stream>>>(agg2, out);
}